// URPE_53206054863260
// MI455X (gfx1250) — compile-verified
//
#include <hip/hip_runtime.h>
#include <stdint.h>

// Problem constants from the reference
#define NB      2
#define NH      16
#define LSEQ    2048
#define VALS    4095        // 2L - 1
#define ROWS    16          // rows of the LxL slice per block
#define THREADS 256

// CDNA5 async global->LDS path (gfx1250). Guarded so the build always succeeds;
// fallback is a plain vectorized LDS fill.
#if defined(__HIP_DEVICE_COMPILE__) && \
    __has_builtin(__builtin_amdgcn_global_load_async_to_lds_b128) && \
    __has_builtin(__builtin_amdgcn_s_wait_asynccnt)
#define USE_ASYNC_LDS 1
#else
#define USE_ASYNC_LDS 0
#endif

#if USE_ASYNC_LDS
typedef int v4i __attribute__((ext_vector_type(4)));
typedef __attribute__((address_space(1))) v4i* gas_v4i_p;   // global int4*
typedef __attribute__((address_space(3))) v4i* lds_v4i_p;   // LDS int4*
#endif

__global__ __launch_bounds__(THREADS)
void urpe_toeplitz_mul(const float* __restrict__ attn,
                       const float* __restrict__ urpe,
                       float* __restrict__ out)
{
    // Raw urpe row for this head: [c(0:2048) | r(0:2048)] = 16 KiB in LDS.
    __shared__ float s_w[2 * LSEQ];

    const int tilesPerSeq = LSEQ / ROWS;                 // 128
    const int tile = blockIdx.x % tilesPerSeq;
    const int h    = (blockIdx.x / tilesPerSeq) % NH;
    const int b    =  blockIdx.x / (tilesPerSeq * NH);
    const int row0 = tile * ROWS;

    // ---- Stage urpe[h, 0:4096] into LDS ----
    const float* wrow = urpe + (size_t)h * (2 * LSEQ);
#if USE_ASYNC_LDS
    for (int t = threadIdx.x; t < (2 * LSEQ) / 4; t += THREADS) {
        // 16B per lane, async DMA into LDS; generic flat addr == global addr.
        __builtin_amdgcn_global_load_async_to_lds_b128(
            (gas_v4i_p)(uintptr_t)(wrow + 4 * t),
            (lds_v4i_p)&s_w[4 * t],
            /*imm offset*/ 0, /*cpol*/ 0);
    }
    __builtin_amdgcn_s_wait_asynccnt(0);   // this wave's async LDS writes done
#else
    for (int t = threadIdx.x; t < (2 * LSEQ) / 4; t += THREADS) {
        const float4 v = *(const float4*)(wrow + 4 * t);
        *(float4*)&s_w[4 * t] = v;
    }
#endif
    __syncthreads();                        // all waves' fills visible

    // ---- Stream 16 rows x 2048 cols of attention_probs ----
    const size_t base = (((size_t)b * NH + h) * LSEQ + (size_t)row0) * LSEQ;
    const float* __restrict__ src = attn + base;
    float*       __restrict__ dst = out  + base;

    for (int r = 0; r < ROWS; ++r) {
        const int i = row0 + r;
        const float4* __restrict__ srow = (const float4*)(src + (size_t)r * LSEQ);
        float4*       __restrict__ drow = (float4*)(dst + (size_t)r * LSEQ);
        for (int c4 = threadIdx.x; c4 < LSEQ / 4; c4 += THREADS) {
            float4 a = srow[c4];            // global_load_b128
            const int j = c4 * 4;
            float w[4];
#pragma unroll
            for (int e = 0; e < 4; ++e) {
                int d = j + e - i;          // in (-L, L)
                d += (d >> 31) & VALS;      // d mod (2L-1), branch-free
                // vals[d] = d < L ? urpe[L+d] : urpe[2L-1-d]
                const int k = (d < LSEQ) ? (LSEQ + d) : (VALS - d);
                w[e] = s_w[k];              // ds_load_b32
            }
            a.x *= w[0];
            a.y *= w[1];
            a.z *= w[2];
            a.w *= w[3];
            drow[c4] = a;                   // global_store_b128
        }
    }
}

extern "C" void kernel_launch(void* const* d_in, const int* in_sizes, int n_in,
                              void* d_out, int out_size, void* d_ws, size_t ws_size,
                              hipStream_t stream)
{
    (void)in_sizes; (void)n_in; (void)out_size; (void)d_ws; (void)ws_size;
    const float* attn = (const float*)d_in[0];   // (2, 16, 2048, 2048) f32
    const float* urpe = (const float*)d_in[1];   // (16, 4096) f32
    float* out = (float*)d_out;                  // (2, 16, 2048, 2048) f32

    const int grid = NB * NH * (LSEQ / ROWS);    // 4096 blocks
    urpe_toeplitz_mul<<<grid, THREADS, 0, stream>>>(attn, urpe, out);
}